// GriffinLim_34119220199742
// MI455X (gfx1250) — compile-verified
//
#include <hip/hip_runtime.h>
#include <math.h>

typedef __attribute__((ext_vector_type(2))) float v2f;
typedef __attribute__((ext_vector_type(8))) float v8f;

namespace {
constexpr int kNFFT  = 1024;
constexpr int kHop   = 256;
constexpr int kMels  = 80;
constexpr int kNF    = 513;
constexpr int kB     = 8;
constexpr int kT     = 1024;
constexpr int kPad   = 512;
constexpr int kLPad  = kNFFT + (kT - 1) * kHop;   // 262912
constexpr int kLOut  = (kT - 1) * kHop;           // 261888
constexpr int kIters = 60;

// table block layout (float offsets)
constexpr int TAB_FFR  = 0;      // forward DFT32 real        cos(2pi ab/32)
constexpr int TAB_FFI  = 1024;   // forward DFT32 imag       -sin
constexpr int TAB_FFNI = 2048;   // forward DFT32 -imag      +sin
constexpr int TAB_GIR  = 3072;   // inverse DFT32 real        cos
constexpr int TAB_GII  = 4096;   // inverse DFT32 imag       +sin
constexpr int TAB_GINI = 5120;   // inverse DFT32 -imag      -sin
constexpr int TAB_TFR  = 6144;   // forward twiddle real
constexpr int TAB_TFI  = 7168;   // forward twiddle imag
constexpr int TAB_TVR  = 8192;   // inverse twiddle real / 1024
constexpr int TAB_TVI  = 9216;   // inverse twiddle imag / 1024
constexpr int TAB_WIN  = 10240;  // hann window
constexpr int TAB_SIZE = 11264;

// workspace layout (float offsets)
constexpr size_t OFF_TABS = 0;
constexpr size_t OFF_ENVR = OFF_TABS + TAB_SIZE;                       // 262912
constexpr size_t OFF_EMEL = OFF_ENVR + kLPad;                         // 655360
constexpr size_t OFF_MAG  = OFF_EMEL + (size_t)kB * kMels * kT;       // 4202496
constexpr size_t OFF_SR   = OFF_MAG  + (size_t)kB * kNF * kT;
constexpr size_t OFF_SI   = OFF_SR   + (size_t)kB * kNF * kT;
constexpr size_t OFF_FRM  = OFF_SI   + (size_t)kB * kNF * kT;         // 8388608
constexpr size_t OFF_Y    = OFF_FRM  + (size_t)kB * kT * kNFFT;       // 2095104
}  // namespace

// ---------------------------------------------------------------------------
// WMMA helpers: V_WMMA_F32_16X16X4_F32, D = A(16x4) * B(4x16) + C(16x16)
// A layout : lane L (m = L%16, h = L/16): v0 = A[m][2h], v1 = A[m][2h+1]
// B layout : lane L:                      v0 = B[2h][m], v1 = B[2h+1][m]
// C/D      : VGPR r: lanes0-15 -> D[r][m], lanes16-31 -> D[r+8][m]
// ---------------------------------------------------------------------------
__device__ __forceinline__ v8f wmma4(v2f a, v2f b, v8f c) {
  return __builtin_amdgcn_wmma_f32_16x16x4_f32(false, a, false, b, (short)0, c,
                                               false, false);
}

__device__ __forceinline__ v2f fragA(const float (*M)[33], int ti, int kc,
                                     int m, int h) {
  int k0 = 4 * kc + 2 * h;
  v2f r;
  r.x = M[16 * ti + m][k0];
  r.y = M[16 * ti + m][k0 + 1];
  return r;
}

__device__ __forceinline__ v2f fragB(const float (*M)[33], int kc, int tj,
                                     int m, int h) {
  int k0 = 4 * kc + 2 * h;
  v2f r;
  r.x = M[k0][16 * tj + m];
  r.y = M[k0 + 1][16 * tj + m];
  return r;
}

// ---------------------------------------------------------------------------
// Table generation (deterministic, cheap, once per launch)
// ---------------------------------------------------------------------------
__global__ void k_tables(float* __restrict__ tab) {
  int t = threadIdx.x;  // 0..1023
  int a = t >> 5, b = t & 31;
  const double two_pi = 6.283185307179586476925287;
  double x32 = two_pi * (double)(a * b) / 32.0;
  double x1k = two_pi * (double)(a * b) / 1024.0;
  float c32 = (float)cos(x32), s32 = (float)sin(x32);
  float c1k = (float)cos(x1k), s1k = (float)sin(x1k);
  tab[TAB_FFR + t]  = c32;
  tab[TAB_FFI + t]  = -s32;
  tab[TAB_FFNI + t] = s32;
  tab[TAB_GIR + t]  = c32;
  tab[TAB_GII + t]  = s32;
  tab[TAB_GINI + t] = -s32;
  tab[TAB_TFR + t]  = c1k;
  tab[TAB_TFI + t]  = -s1k;
  tab[TAB_TVR + t]  = c1k * (1.0f / 1024.0f);
  tab[TAB_TVI + t]  = s1k * (1.0f / 1024.0f);
  tab[TAB_WIN + t]  = 0.5f - 0.5f * (float)cos(two_pi * (double)t / 1024.0);
}

// envelope reciprocal: env[p] = sum_t win[p-256t]^2, recip with 1e-11 guard
__global__ void k_env(const float* __restrict__ tab, float* __restrict__ envr) {
  int p = blockIdx.x * blockDim.x + threadIdx.x;
  if (p >= kLPad) return;
  const float* win = tab + TAB_WIN;
  int t1 = p >> 8;             if (t1 > kT - 1) t1 = kT - 1;
  int t0 = (p - 768) >> 8;     if (t0 < 0) t0 = 0;  // ceil((p-1023)/256)
  float e = 0.f;
  for (int t = t0; t <= t1; ++t) {
    float w = win[p - 256 * t];
    e += w * w;
  }
  envr[p] = (e > 1e-11f) ? 1.0f / e : 1.0f;
}

__global__ void k_expmel(const float* __restrict__ mel, float* __restrict__ em,
                         int n) {
  int i = blockIdx.x * blockDim.x + threadIdx.x;
  if (i < n) em[i] = __expf(mel[i]);
}

// mag[b][f][t] = max(0, sum_m inv_fb[f][m] * exp(mel[b][m][t]))
__global__ void k_mag(const float* __restrict__ inv_fb,
                      const float* __restrict__ em, float* __restrict__ mag) {
  int i = blockIdx.x * blockDim.x + threadIdx.x;
  if (i >= kB * kNF * kT) return;
  int t = i & (kT - 1);
  int rem = i >> 10;            // b*kNF + f
  int f = rem % kNF, b = rem / kNF;
  const float* fb = inv_fb + (size_t)f * kMels;
  const float* e = em + ((size_t)b * kMels) * kT + t;
  float acc = 0.f;
#pragma unroll 8
  for (int m = 0; m < kMels; ++m) acc += fb[m] * e[(size_t)m * kT];
  mag[i] = fmaxf(acc, 0.f);
}

__global__ void k_specinit(const float* __restrict__ angles,
                           const float* __restrict__ mag,
                           float* __restrict__ sr, float* __restrict__ si) {
  int i = blockIdx.x * blockDim.x + threadIdx.x;
  if (i >= kB * kNF * kT) return;
  float s, c;
  __sincosf(6.2831853071795864f * angles[i], &s, &c);
  float m = mag[i];
  sr[i] = m * c;
  si[i] = m * s;
}

// ---------------------------------------------------------------------------
// iSTFT frames: per wave, one frame.  1024-pt inverse FFT of the Hermitian
// extension of spec[b,:,t], via 32x32 four-step on V_WMMA_F32_16X16X4_F32.
// frames[b][t][j] = Re(x[j]) * win[j]   (1/1024 folded into inverse twiddle)
// ---------------------------------------------------------------------------
__global__ void __launch_bounds__(128)
k_istft(const float* __restrict__ specR, const float* __restrict__ specI,
        const float* __restrict__ tab, float* __restrict__ frames) {
  __shared__ float sBuf[4][2][32][33];
  __shared__ float sG[3][32][33];
  __shared__ float sTw[2][32][33];
  __shared__ float sWin[1024];

  for (int i = threadIdx.x; i < 1024; i += 128) {
    int r = i >> 5, c = i & 31;
    sG[0][r][c]  = tab[TAB_GIR + i];
    sG[1][r][c]  = tab[TAB_GII + i];
    sG[2][r][c]  = tab[TAB_GINI + i];
    sTw[0][r][c] = tab[TAB_TVR + i];
    sTw[1][r][c] = tab[TAB_TVI + i];
    sWin[i]      = tab[TAB_WIN + i];
  }

  const int w = threadIdx.x >> 5, lane = threadIdx.x & 31;
  const int m = lane & 15, h = lane >> 4;
  const int frame = blockIdx.x * 4 + w;   // 0..8191
  const int b = frame >> 10, t = frame & 1023;
  float (*Ar)[33] = sBuf[w][0];
  float (*Ai)[33] = sBuf[w][1];
  const float (*G0)[33] = sG[0];
  const float (*G1)[33] = sG[1];
  const float (*G2)[33] = sG[2];

  // load Hermitian spectrum: A[k1][k2] = Z[k1 + 32*k2]
  const float* sR = specR + ((size_t)b * kNF) * kT + t;
  const float* sI = specI + ((size_t)b * kNF) * kT + t;
  for (int i = lane; i < 1024; i += 32) {
    int k1 = i >> 5, k2 = i & 31;
    int k = k1 + 32 * k2;
    float zr, zi;
    if (k <= 512) {
      zr = sR[(size_t)k * kT];
      zi = sI[(size_t)k * kT];
    } else {
      int kk = 1024 - k;
      zr = sR[(size_t)kk * kT];
      zi = -sI[(size_t)kk * kT];
    }
    Ar[k1][k2] = zr;
    Ai[k1][k2] = zi;
  }
  __syncthreads();

  // stage 1: Y = A x G  (complex x complex)
  v8f Yr[2][2], Yi[2][2];
#pragma unroll
  for (int ti = 0; ti < 2; ++ti)
#pragma unroll
    for (int tj = 0; tj < 2; ++tj) { Yr[ti][tj] = {}; Yi[ti][tj] = {}; }
#pragma unroll
  for (int kc = 0; kc < 8; ++kc) {
#pragma unroll
    for (int ti = 0; ti < 2; ++ti) {
      v2f ar = fragA(Ar, ti, kc, m, h);
      v2f ai = fragA(Ai, ti, kc, m, h);
#pragma unroll
      for (int tj = 0; tj < 2; ++tj) {
        v2f gr  = fragB(G0, kc, tj, m, h);
        v2f gi  = fragB(G1, kc, tj, m, h);
        v2f gni = fragB(G2, kc, tj, m, h);
        Yr[ti][tj] = wmma4(ar, gr, Yr[ti][tj]);
        Yr[ti][tj] = wmma4(ai, gni, Yr[ti][tj]);
        Yi[ti][tj] = wmma4(ar, gi, Yi[ti][tj]);
        Yi[ti][tj] = wmma4(ai, gr, Yi[ti][tj]);
      }
    }
  }
  __syncthreads();

  // twiddle (includes 1/1024) and write back into A
#pragma unroll
  for (int ti = 0; ti < 2; ++ti)
#pragma unroll
    for (int tj = 0; tj < 2; ++tj)
#pragma unroll
      for (int r = 0; r < 8; ++r) {
        int row = 16 * ti + r + 8 * h, col = 16 * tj + m;
        float tr = sTw[0][row][col], tw = sTw[1][row][col];
        float yr = Yr[ti][tj][r], yi = Yi[ti][tj][r];
        Ar[row][col] = yr * tr - yi * tw;
        Ai[row][col] = yr * tw + yi * tr;
      }
  __syncthreads();

  // stage 2: X = G x Y'  -- only the real part is needed
  v8f Xr[2][2];
#pragma unroll
  for (int ti = 0; ti < 2; ++ti)
#pragma unroll
    for (int tj = 0; tj < 2; ++tj) Xr[ti][tj] = {};
#pragma unroll
  for (int kc = 0; kc < 8; ++kc) {
#pragma unroll
    for (int ti = 0; ti < 2; ++ti) {
      v2f gr  = fragA(G0, ti, kc, m, h);
      v2f gni = fragA(G2, ti, kc, m, h);
#pragma unroll
      for (int tj = 0; tj < 2; ++tj) {
        v2f yr = fragB(Ar, kc, tj, m, h);
        v2f yi = fragB(Ai, kc, tj, m, h);
        Xr[ti][tj] = wmma4(gr, yr, Xr[ti][tj]);
        Xr[ti][tj] = wmma4(gni, yi, Xr[ti][tj]);
      }
    }
  }
  __syncthreads();
#pragma unroll
  for (int ti = 0; ti < 2; ++ti)
#pragma unroll
    for (int tj = 0; tj < 2; ++tj)
#pragma unroll
      for (int r = 0; r < 8; ++r)
        Ar[16 * ti + r + 8 * h][16 * tj + m] = Xr[ti][tj][r];
  __syncthreads();

  // x[j] = X_mat[j>>5][j&31]; apply synthesis window
  float* fo = frames + (size_t)frame * kNFFT;
  for (int i = lane; i < 1024; i += 32)
    fo[i] = Ar[i >> 5][i & 31] * sWin[i];
}

// overlap-add + envelope normalization; gather formulation (<=4 frames/sample)
__global__ void k_ola(const float* __restrict__ frames,
                      const float* __restrict__ envr, float* __restrict__ y) {
  int i = blockIdx.x * blockDim.x + threadIdx.x;
  if (i >= kB * kLOut) return;
  int b = i / kLOut, l = i - b * kLOut;
  int p = l + kPad;
  int t1 = p >> 8;          if (t1 > kT - 1) t1 = kT - 1;
  int t0 = (p - 768) >> 8;  if (t0 < 0) t0 = 0;
  const float* fb = frames + ((size_t)b * kT) * kNFFT;
  float acc = 0.f;
  for (int t = t0; t <= t1; ++t) acc += fb[(size_t)t * kNFFT + (p - 256 * t)];
  y[i] = acc * envr[p];
}

// ---------------------------------------------------------------------------
// STFT frames + Griffin-Lim phase projection: gather reflect-padded frame,
// window, forward 1024-pt FFT via WMMA four-step, spec = mag * s/(|s|+1e-8)
// ---------------------------------------------------------------------------
__global__ void __launch_bounds__(128)
k_stft(const float* __restrict__ y, const float* __restrict__ tab,
       const float* __restrict__ mag, float* __restrict__ specR,
       float* __restrict__ specI) {
  __shared__ float sBuf[4][2][32][33];
  __shared__ float sF[3][32][33];
  __shared__ float sTw[2][32][33];
  __shared__ float sWin[1024];

  for (int i = threadIdx.x; i < 1024; i += 128) {
    int r = i >> 5, c = i & 31;
    sF[0][r][c]  = tab[TAB_FFR + i];
    sF[1][r][c]  = tab[TAB_FFI + i];
    sF[2][r][c]  = tab[TAB_FFNI + i];
    sTw[0][r][c] = tab[TAB_TFR + i];
    sTw[1][r][c] = tab[TAB_TFI + i];
    sWin[i]      = tab[TAB_WIN + i];
  }

  const int w = threadIdx.x >> 5, lane = threadIdx.x & 31;
  const int m = lane & 15, h = lane >> 4;
  const int frame = blockIdx.x * 4 + w;
  const int b = frame >> 10, t = frame & 1023;
  float (*Ar)[33] = sBuf[w][0];
  float (*Ai)[33] = sBuf[w][1];
  const float (*F0)[33] = sF[0];
  const float (*F1)[33] = sF[1];
  const float (*F2)[33] = sF[2];

  // gather windowed frame with reflect padding: A[n1][n2] = xw[n1 + 32*n2]
  const float* yb = y + (size_t)b * kLOut;
  for (int i = lane; i < 1024; i += 32) {
    int n1 = i >> 5, n2 = i & 31;
    int n = n1 + 32 * n2;
    int p = 256 * t + n - kPad;
    if (p < 0) p = -p;
    else if (p >= kLOut) p = 2 * kLOut - 2 - p;
    Ar[n1][n2] = yb[p] * sWin[n];
  }
  __syncthreads();

  // stage 1 (real input): Yr = A*Fr, Yi = A*Fi
  v8f Yr[2][2], Yi[2][2];
#pragma unroll
  for (int ti = 0; ti < 2; ++ti)
#pragma unroll
    for (int tj = 0; tj < 2; ++tj) { Yr[ti][tj] = {}; Yi[ti][tj] = {}; }
#pragma unroll
  for (int kc = 0; kc < 8; ++kc) {
#pragma unroll
    for (int ti = 0; ti < 2; ++ti) {
      v2f ar = fragA(Ar, ti, kc, m, h);
#pragma unroll
      for (int tj = 0; tj < 2; ++tj) {
        v2f fr = fragB(F0, kc, tj, m, h);
        v2f fi = fragB(F1, kc, tj, m, h);
        Yr[ti][tj] = wmma4(ar, fr, Yr[ti][tj]);
        Yi[ti][tj] = wmma4(ar, fi, Yi[ti][tj]);
      }
    }
  }
  __syncthreads();

#pragma unroll
  for (int ti = 0; ti < 2; ++ti)
#pragma unroll
    for (int tj = 0; tj < 2; ++tj)
#pragma unroll
      for (int r = 0; r < 8; ++r) {
        int row = 16 * ti + r + 8 * h, col = 16 * tj + m;
        float tr = sTw[0][row][col], tw = sTw[1][row][col];
        float yr = Yr[ti][tj][r], yi = Yi[ti][tj][r];
        Ar[row][col] = yr * tr - yi * tw;
        Ai[row][col] = yr * tw + yi * tr;
      }
  __syncthreads();

  // stage 2: X = F x Y' (full complex)
  v8f Xr[2][2], Xi[2][2];
#pragma unroll
  for (int ti = 0; ti < 2; ++ti)
#pragma unroll
    for (int tj = 0; tj < 2; ++tj) { Xr[ti][tj] = {}; Xi[ti][tj] = {}; }
#pragma unroll
  for (int kc = 0; kc < 8; ++kc) {
#pragma unroll
    for (int ti = 0; ti < 2; ++ti) {
      v2f fr  = fragA(F0, ti, kc, m, h);
      v2f fi  = fragA(F1, ti, kc, m, h);
      v2f fni = fragA(F2, ti, kc, m, h);
#pragma unroll
      for (int tj = 0; tj < 2; ++tj) {
        v2f yr = fragB(Ar, kc, tj, m, h);
        v2f yi = fragB(Ai, kc, tj, m, h);
        Xr[ti][tj] = wmma4(fr, yr, Xr[ti][tj]);
        Xr[ti][tj] = wmma4(fni, yi, Xr[ti][tj]);
        Xi[ti][tj] = wmma4(fr, yi, Xi[ti][tj]);
        Xi[ti][tj] = wmma4(fi, yr, Xi[ti][tj]);
      }
    }
  }
  __syncthreads();
#pragma unroll
  for (int ti = 0; ti < 2; ++ti)
#pragma unroll
    for (int tj = 0; tj < 2; ++tj)
#pragma unroll
      for (int r = 0; r < 8; ++r) {
        int row = 16 * ti + r + 8 * h, col = 16 * tj + m;
        Ar[row][col] = Xr[ti][tj][r];
        Ai[row][col] = Xi[ti][tj][r];
      }
  __syncthreads();

  // phase projection on bins 0..512:  spec = mag * s / (|s| + 1e-8)
  const size_t base = ((size_t)b * kNF) * kT + t;
  for (int k = lane; k < kNF; k += 32) {
    float xr = Ar[k >> 5][k & 31];
    float xi = Ai[k >> 5][k & 31];
    float inv = mag[base + (size_t)k * kT] /
                (__fsqrt_rn(xr * xr + xi * xi) + 1e-8f);
    specR[base + (size_t)k * kT] = xr * inv;
    specI[base + (size_t)k * kT] = xi * inv;
  }
}

// ---------------------------------------------------------------------------
extern "C" void kernel_launch(void* const* d_in, const int* in_sizes, int n_in,
                              void* d_out, int out_size, void* d_ws,
                              size_t ws_size, hipStream_t stream) {
  (void)in_sizes; (void)n_in; (void)out_size; (void)ws_size;
  const float* mel    = (const float*)d_in[0];  // (B, 80, T)
  const float* angles = (const float*)d_in[1];  // (B, 513, T)
  const float* inv_fb = (const float*)d_in[2];  // (513, 80)
  float* out = (float*)d_out;                   // (B, 261888)
  float* ws  = (float*)d_ws;

  float* tabs   = ws + OFF_TABS;
  float* envr   = ws + OFF_ENVR;
  float* emel   = ws + OFF_EMEL;
  float* mag    = ws + OFF_MAG;
  float* specR  = ws + OFF_SR;
  float* specI  = ws + OFF_SI;
  float* frames = ws + OFF_FRM;
  float* ybuf   = ws + OFF_Y;

  constexpr int nSpec = kB * kNF * kT;
  constexpr int nMel  = kB * kMels * kT;
  constexpr int nY    = kB * kLOut;
  constexpr int fftBlocks = (kB * kT) / 4;  // 4 frames (waves) per block

  k_tables<<<1, 1024, 0, stream>>>(tabs);
  k_env<<<(kLPad + 255) / 256, 256, 0, stream>>>(tabs, envr);
  k_expmel<<<(nMel + 255) / 256, 256, 0, stream>>>(mel, emel, nMel);
  k_mag<<<(nSpec + 255) / 256, 256, 0, stream>>>(inv_fb, emel, mag);
  k_specinit<<<(nSpec + 255) / 256, 256, 0, stream>>>(angles, mag, specR,
                                                      specI);
  for (int it = 0; it < kIters; ++it) {
    k_istft<<<fftBlocks, 128, 0, stream>>>(specR, specI, tabs, frames);
    k_ola<<<(nY + 255) / 256, 256, 0, stream>>>(frames, envr, ybuf);
    k_stft<<<fftBlocks, 128, 0, stream>>>(ybuf, tabs, mag, specR, specI);
  }
  k_istft<<<fftBlocks, 128, 0, stream>>>(specR, specI, tabs, frames);
  k_ola<<<(nY + 255) / 256, 256, 0, stream>>>(frames, envr, out);
}